// WindowMasking_77043123356180
// MI455X (gfx1250) — compile-verified
//
#include <hip/hip_runtime.h>
#include <stdint.h>

// Problem constants (from the reference): x is (2,3,224,224) f32,
// KERNEL=STRIDE=16 -> 14x14 windows (p=196), mask = all-true (exact tiling).
#define N_    2
#define C_    3
#define H_    224
#define W_    224
#define KW    16
#define SW    16
#define NWIN  14                     // windows per spatial dim
#define P_    (NWIN * NWIN)          // 196
#define HW    (H_ * W_)              // 50176 floats per plane
#define ROWS  16                     // rows staged per block
#define CHUNK (ROWS * W_)            // 3584 floats = 14336 B of LDS
#define CHUNK_V4 (CHUNK / 4)         // 896 float4
#define CHUNKS_PER_PLANE (H_ / ROWS) // 14
#define WGROUPS 14                   // window-groups per chunk
#define WIN_PER_GROUP (P_ / WGROUPS) // 14 windows written per block
#define THREADS 224                  // 7 wave32 waves; 896 % 224 == 0 (exact trips)
#define V4_PER_THREAD (CHUNK_V4 / THREADS) // 4
#define NBLOCKS (N_ * C_ * CHUNKS_PER_PLANE * WGROUPS) // 1176

typedef float v4f __attribute__((ext_vector_type(4)));  // true clang vector (B128)

__global__ __launch_bounds__(THREADS)
void WindowMasking_bcast_kernel(const float* __restrict__ x,
                                float* __restrict__ out) {
    __shared__ float smem[CHUNK];

    const int tid     = threadIdx.x;
    const int wg      = blockIdx.x % WGROUPS;          // which group of 14 windows
    const int chunkId = blockIdx.x / WGROUPS;
    const int plane   = chunkId / CHUNKS_PER_PLANE;    // n*C + c  (0..5)
    const int cip     = chunkId % CHUNKS_PER_PLANE;    // chunk index within plane
    const int n       = plane / C_;
    const int c       = plane % C_;
    const int row0    = cip * ROWS;

    const float* __restrict__ src = x + (size_t)plane * HW + (size_t)row0 * W_;

    // ---- Stage chunk Global -> LDS via CDNA5 async path (ASYNCcnt) ----------
    // Exact 4 iterations per thread, lane-contiguous B128 transfers.
#pragma unroll
    for (int k = 0; k < V4_PER_THREAD; ++k) {
        const int i = tid + k * THREADS;               // v4f index
        // Low 32 bits of a generic shared-memory address ARE the LDS byte
        // offset (aperture lives in addr[63:32]); hw adds it to the wave's LDS base.
        uint32_t lds_off = (uint32_t)(uintptr_t)(&smem[i * 4]);
        uint64_t gaddr   = (uint64_t)(uintptr_t)(src + i * 4);
        asm volatile("global_load_async_to_lds_b128 %0, %1, off"
                     :: "v"(lds_off), "v"(gaddr)
                     : "memory");
    }
#if __has_builtin(__builtin_amdgcn_s_wait_asynccnt)
    __builtin_amdgcn_s_wait_asynccnt(0);
#else
    asm volatile("s_wait_asynccnt 0x0" ::: "memory");
#endif
    __syncthreads();

    // ---- Faithful coverage mask (constant-folds to all-true for 224/16/16) --
    for (int idx = tid; idx < CHUNK; idx += THREADS) { // exactly 16 trips
        int h  = row0 + idx / W_;
        int w  = idx % W_;
        int gh = h / SW; if (gh > NWIN - 1) gh = NWIN - 1;
        int gw = w / SW; if (gw > NWIN - 1) gw = NWIN - 1;
        bool covered = ((h - gh * SW) < KW) && ((w - gw * SW) < KW);
        if (!covered) smem[idx] = 0.0f;
    }
    __syncthreads();

    // ---- Pull this thread's 64 B into VGPRs once (single LDS read) ----------
    v4f r[V4_PER_THREAD];
#pragma unroll
    for (int k = 0; k < V4_PER_THREAD; ++k)
        r[k] = *(const v4f*)&smem[(tid + k * THREADS) * 4];

    // ---- Broadcast to 14 window slots: 56 streaming NT B128 stores ----------
    // Fully unrolled: max byte offset 13*602112 + 3*3584 = 7.84 MB fits the
    // signed 24-bit IOFFSET, so all stores share one base VGPR pair and use
    // immediate offsets (zero per-store address math, deep store queue).
    float* __restrict__ dst0 =
        out + (((size_t)(n * P_ + wg * WIN_PER_GROUP)) * C_ + c) * HW
            + (size_t)row0 * W_ + (size_t)tid * 4;
#pragma unroll
    for (int wi = 0; wi < WIN_PER_GROUP; ++wi) {
        float* __restrict__ dst = dst0 + (size_t)wi * (C_ * HW);
#pragma unroll
        for (int k = 0; k < V4_PER_THREAD; ++k)
            __builtin_nontemporal_store(r[k], (v4f*)(dst + k * THREADS * 4));
    }
}

extern "C" void kernel_launch(void* const* d_in, const int* in_sizes, int n_in,
                              void* d_out, int out_size, void* d_ws, size_t ws_size,
                              hipStream_t stream) {
    (void)in_sizes; (void)n_in; (void)out_size; (void)d_ws; (void)ws_size;
    const float* x = (const float*)d_in[0];
    float* out     = (float*)d_out;
    WindowMasking_bcast_kernel<<<dim3(NBLOCKS), dim3(THREADS), 0, stream>>>(x, out);
}